// RmiModel_34514357191210
// MI455X (gfx1250) — compile-verified
//
#include <hip/hip_runtime.h>

// Reference shapes: x (B,7,N) f32, calib (6,6), bias (6,1); out (B,15) f32.
#define NSAMP 200
#define NSEG  8     // segments per batch for the parallel scan (8 waves/SIMD)

typedef float v2f __attribute__((ext_vector_type(2)));
typedef float v8f __attribute__((ext_vector_type(8)));

// ---------------------------------------------------------------------------
// One preintegration step (per-thread):
//   Ca = C@a;  P += V dt + 0.5 Ca dt^2;  V += Ca dt;  C = C @ so3_exp(w dt)
// ---------------------------------------------------------------------------
__device__ __forceinline__ void integrate_step(
    float P[3], float V[3], float C[9],
    float wx, float wy, float wz,
    float ax, float ay, float az, float dt)
{
    float Cax = C[0]*ax + C[1]*ay + C[2]*az;
    float Cay = C[3]*ax + C[4]*ay + C[5]*az;
    float Caz = C[6]*ax + C[7]*ay + C[8]*az;
    float hdt2 = 0.5f * dt * dt;
    P[0] += V[0]*dt + Cax*hdt2;
    P[1] += V[1]*dt + Cay*hdt2;
    P[2] += V[2]*dt + Caz*hdt2;
    V[0] += Cax*dt; V[1] += Cay*dt; V[2] += Caz*dt;

    float px = wx*dt, py = wy*dt, pz = wz*dt;
    float t2 = px*px + py*py + pz*pz;
    float A, Bc;
    if (t2 < 1e-8f) {
        A  = 1.0f - t2 * (1.0f/6.0f);
        Bc = 0.5f - t2 * (1.0f/24.0f);
    } else {
        float th = sqrtf(t2);
        float s, c;
        __sincosf(th, &s, &c);
        A  = __fdividef(s, th);
        Bc = __fdividef(1.0f - c, t2);
    }
    float xx=px*px, yy=py*py, zz=pz*pz;
    float xy=px*py, xz=px*pz, yz=py*pz;
    float R00 = 1.0f - Bc*(yy+zz);
    float R01 = Bc*xy - A*pz;
    float R02 = Bc*xz + A*py;
    float R10 = Bc*xy + A*pz;
    float R11 = 1.0f - Bc*(xx+zz);
    float R12 = Bc*yz - A*px;
    float R20 = Bc*xz - A*py;
    float R21 = Bc*yz + A*px;
    float R22 = 1.0f - Bc*(xx+yy);

    float c0 = C[0]*R00 + C[1]*R10 + C[2]*R20;
    float c1 = C[0]*R01 + C[1]*R11 + C[2]*R21;
    float c2 = C[0]*R02 + C[1]*R12 + C[2]*R22;
    float c3 = C[3]*R00 + C[4]*R10 + C[5]*R20;
    float c4 = C[3]*R01 + C[4]*R11 + C[5]*R21;
    float c5 = C[3]*R02 + C[4]*R12 + C[5]*R22;
    float c6 = C[6]*R00 + C[7]*R10 + C[8]*R20;
    float c7 = C[6]*R01 + C[7]*R11 + C[8]*R21;
    float c8 = C[6]*R02 + C[7]*R12 + C[8]*R22;
    C[0]=c0;C[1]=c1;C[2]=c2;C[3]=c3;C[4]=c4;C[5]=c5;C[6]=c6;C[7]=c7;C[8]=c8;
}

// ---------------------------------------------------------------------------
// Kernel 1: calibration GEMM via V_WMMA_F32_16X16X4_F32.
// One wave handles one (batch, 16-sample tile):
//   D(16x16) = A(16x4: rows of M6=I+calib, K=0..3) x B(4x16: input chans as
//   K-rows, samples as cols) + [first WMMA for K=4..5, zero padded] + bias(C)
// Writes y to ws in [channel][n][batch] layout (scan kernel reads coalesced),
// channel 6 = dt = t[n+1]-t[n].
// A layout (32-bit 16x4): lanes 0-15 row M=lane, VGPR0/1 = K0/K1;
//                         lanes 16-31: K2/K3.
// C/D layout: VGPR r = row r for lanes 0-15 (rows 8+r in lanes 16-31, unused).
// ---------------------------------------------------------------------------
__global__ __launch_bounds__(128)
void calib_wmma_kernel(const float* __restrict__ x,
                       const float* __restrict__ calib,
                       const float* __restrict__ bias,
                       float* __restrict__ ws_y,
                       int B)
{
    const int NT = (NSAMP + 15) / 16;                 // 13 tiles per batch
    int wave = blockIdx.x * (blockDim.x >> 5) + (threadIdx.x >> 5);
    if (wave >= B * NT) return;
    int b  = wave / NT;
    int n0 = (wave - b * NT) * 16;
    int lane = threadIdx.x & 31;
    int half = lane >> 4;      // which K-pair this lane supplies
    int l    = lane & 15;      // row (A) / column-sample (B)

    const float* xb = x + (size_t)b * 7 * NSAMP;

    // M6 = I + calib, zero-padded to 16 rows x 8 cols
    auto m6 = [&](int r, int c) -> float {
        if (r < 6 && c < 6) {
            float v = calib[r * 6 + c];
            return (r == c) ? (v + 1.0f) : v;
        }
        return 0.0f;
    };
    int kb1 = half * 2;        // second WMMA covers K = 0..3
    int kb2 = 4 + half * 2;    // first WMMA covers K = 4..5 (6,7 are zero)
    v2f a1; a1.x = m6(l, kb1); a1.y = m6(l, kb1 + 1);
    v2f a2; a2.x = m6(l, kb2); a2.y = m6(l, kb2 + 1);

    // B operand: K-row k = input channel (1+k), column = sample n0+l
    int  n   = n0 + l;
    bool nin = (n < NSAMP);
    auto xr = [&](int k) -> float {
        return (nin && k < 6) ? xb[(size_t)(1 + k) * NSAMP + n] : 0.0f;
    };
    v2f b1; b1.x = xr(kb1); b1.y = xr(kb1 + 1);
    v2f b2; b2.x = xr(kb2); b2.y = xr(kb2 + 1);

    // C operand: bias broadcast along columns, rows 0..5 only
    v8f c = {};
    if (half == 0) {
        #pragma unroll
        for (int r = 0; r < 6; ++r) c[r] = bias[r];
    }

    v8f d = __builtin_amdgcn_wmma_f32_16x16x4_f32(false, a2, false, b2,
                                                  (short)0, c, false, false);
    d     = __builtin_amdgcn_wmma_f32_16x16x4_f32(false, a1, false, b1,
                                                  (short)0, d, false, false);

    if (half == 0 && nin) {
        #pragma unroll
        for (int r = 0; r < 6; ++r)
            ws_y[((size_t)r * NSAMP + n) * B + b] = d[r];
        if (n < NSAMP - 1)   // dt channel
            ws_y[((size_t)6 * NSAMP + n) * B + b] = xb[n + 1] - xb[n];
    }
}

// ---------------------------------------------------------------------------
// Kernel 2a: segmented scan. thread = (seg, batch); batch is the fast index so
// every per-step load of y[ch][n][b] is a fully coalesced 128B line per wave.
// Each segment integrates ~25 steps from identity and stores (P,V,C,T).
// ---------------------------------------------------------------------------
__global__ __launch_bounds__(256)
void seg_scan_kernel(const float* __restrict__ ws_y,
                     float* __restrict__ seg_out, int B)
{
    int tid = blockIdx.x * blockDim.x + threadIdx.x;
    if (tid >= B * NSEG) return;
    int seg = tid / B;
    int b   = tid - seg * B;

    const int SL   = (NSAMP - 1 + NSEG - 1) / NSEG;   // 25
    int nbeg = seg * SL;
    int nend = nbeg + SL;
    if (nend > NSAMP - 1) nend = NSAMP - 1;

    float P[3] = {0.f, 0.f, 0.f};
    float V[3] = {0.f, 0.f, 0.f};
    float C[9] = {1.f,0.f,0.f, 0.f,1.f,0.f, 0.f,0.f,1.f};
    float T = 0.f;

    for (int n = nbeg; n < nend; ++n) {
        size_t col = (size_t)n * B + b;
        // stream ahead (lowers to global_prefetch_b8; speculative, stays in ws)
        __builtin_prefetch(ws_y + (size_t)6 * NSAMP * B + col + (size_t)16 * B, 0, 1);
        float dt = ws_y[(size_t)6 * NSAMP * B + col];
        float wx = ws_y[(size_t)0 * NSAMP * B + col];
        float wy = ws_y[(size_t)1 * NSAMP * B + col];
        float wz = ws_y[(size_t)2 * NSAMP * B + col];
        float ax = ws_y[(size_t)3 * NSAMP * B + col];
        float ay = ws_y[(size_t)4 * NSAMP * B + col];
        float az = ws_y[(size_t)5 * NSAMP * B + col];
        integrate_step(P, V, C, wx, wy, wz, ax, ay, az, dt);
        T += dt;
    }

    float* o = seg_out + (size_t)tid * 16;
    o[0]=P[0]; o[1]=P[1]; o[2]=P[2];
    o[3]=V[0]; o[4]=V[1]; o[5]=V[2];
    #pragma unroll
    for (int k = 0; k < 9; ++k) o[6 + k] = C[k];
    o[15] = T;
}

// ---------------------------------------------------------------------------
// Kernel 2b: compose segments. (P,V,C) o (p,v,c,T) =
//   (P + V*T + C p,  V + C v,  C c)   -- affine group composition.
// ---------------------------------------------------------------------------
__global__ __launch_bounds__(256)
void compose_kernel(const float* __restrict__ seg_out,
                    float* __restrict__ out, int B)
{
    int b = blockIdx.x * blockDim.x + threadIdx.x;
    if (b >= B) return;

    float P[3], V[3], C[9];
    {
        const float* s0 = seg_out + (size_t)b * 16;   // seg 0
        P[0]=s0[0]; P[1]=s0[1]; P[2]=s0[2];
        V[0]=s0[3]; V[1]=s0[4]; V[2]=s0[5];
        #pragma unroll
        for (int k = 0; k < 9; ++k) C[k] = s0[6 + k];
    }
    for (int s = 1; s < NSEG; ++s) {
        const float* ss = seg_out + ((size_t)s * B + b) * 16;
        float p[3] = {ss[0], ss[1], ss[2]};
        float v[3] = {ss[3], ss[4], ss[5]};
        float Ts   = ss[15];
        float Cp[3], Cv[3];
        #pragma unroll
        for (int i = 0; i < 3; ++i) {
            Cp[i] = C[i*3+0]*p[0] + C[i*3+1]*p[1] + C[i*3+2]*p[2];
            Cv[i] = C[i*3+0]*v[0] + C[i*3+1]*v[1] + C[i*3+2]*v[2];
        }
        #pragma unroll
        for (int i = 0; i < 3; ++i) {
            P[i] += V[i]*Ts + Cp[i];
            V[i] += Cv[i];
        }
        float Cn[9];
        #pragma unroll
        for (int i = 0; i < 3; ++i)
            #pragma unroll
            for (int j = 0; j < 3; ++j)
                Cn[i*3+j] = C[i*3+0]*ss[6+0*3+j] + C[i*3+1]*ss[6+1*3+j]
                          + C[i*3+2]*ss[6+2*3+j];
        #pragma unroll
        for (int k = 0; k < 9; ++k) C[k] = Cn[k];
    }

    float* o = out + (size_t)b * 15;
    o[0]=P[0]; o[1]=P[1]; o[2]=P[2];
    o[3]=V[0]; o[4]=V[1]; o[5]=V[2];
    #pragma unroll
    for (int k = 0; k < 9; ++k) o[6 + k] = C[k];
}

// ---------------------------------------------------------------------------
// Fallback: fully fused one-thread-per-batch scan straight from x (used only
// if the workspace is too small for the staged path).
// ---------------------------------------------------------------------------
__global__ __launch_bounds__(256)
void fused_scan_kernel(const float* __restrict__ x,
                       const float* __restrict__ calib,
                       const float* __restrict__ bias,
                       float* __restrict__ out, int B)
{
    int b = blockIdx.x * blockDim.x + threadIdx.x;
    if (b >= B) return;

    float M[36], bi[6];
    #pragma unroll
    for (int r = 0; r < 6; ++r) {
        bi[r] = bias[r];
        #pragma unroll
        for (int c = 0; c < 6; ++c)
            M[r*6+c] = calib[r*6+c] + ((r == c) ? 1.0f : 0.0f);
    }

    const float* xb = x + (size_t)b * 7 * NSAMP;
    float P[3] = {0.f,0.f,0.f}, V[3] = {0.f,0.f,0.f};
    float C[9] = {1.f,0.f,0.f, 0.f,1.f,0.f, 0.f,0.f,1.f};
    float tprev = xb[0];

    for (int n = 0; n < NSAMP - 1; ++n) {
        float tn = xb[n + 1];
        float dt = tn - tprev;
        tprev = tn;
        float xr[6];
        #pragma unroll
        for (int j = 0; j < 6; ++j) xr[j] = xb[(size_t)(1 + j) * NSAMP + n];
        float y[6];
        #pragma unroll
        for (int c = 0; c < 6; ++c) {
            float acc = bi[c];
            #pragma unroll
            for (int j = 0; j < 6; ++j) acc += M[c*6+j] * xr[j];
            y[c] = acc;
        }
        integrate_step(P, V, C, y[0], y[1], y[2], y[3], y[4], y[5], dt);
    }

    float* o = out + (size_t)b * 15;
    o[0]=P[0]; o[1]=P[1]; o[2]=P[2];
    o[3]=V[0]; o[4]=V[1]; o[5]=V[2];
    #pragma unroll
    for (int k = 0; k < 9; ++k) o[6 + k] = C[k];
}

// ---------------------------------------------------------------------------
extern "C" void kernel_launch(void* const* d_in, const int* in_sizes, int n_in,
                              void* d_out, int out_size, void* d_ws, size_t ws_size,
                              hipStream_t stream) {
    const float* x     = (const float*)d_in[0];   // (B,7,N)
    const float* calib = (const float*)d_in[1];   // (6,6)
    const float* bias  = (const float*)d_in[2];   // (6,1)
    float* out = (float*)d_out;                   // (B,15)

    int B = in_sizes[0] / (7 * NSAMP);

    size_t y_floats   = (size_t)7 * NSAMP * B;        // calibrated y + dt, [ch][n][b]
    size_t seg_floats = (size_t)NSEG * B * 16;        // per-segment (P,V,C,T)
    size_t needed     = (y_floats + seg_floats) * sizeof(float);

    if (ws_size >= needed) {
        float* ws_y   = (float*)d_ws;
        float* ws_seg = ws_y + y_floats;

        const int NT = (NSAMP + 15) / 16;
        int waves   = B * NT;
        int blocks1 = (waves + 3) / 4;                // 4 waves / 128-thread block
        calib_wmma_kernel<<<blocks1, 128, 0, stream>>>(x, calib, bias, ws_y, B);

        int t2 = B * NSEG;
        seg_scan_kernel<<<(t2 + 255) / 256, 256, 0, stream>>>(ws_y, ws_seg, B);
        compose_kernel<<<(B + 255) / 256, 256, 0, stream>>>(ws_seg, out, B);
    } else {
        fused_scan_kernel<<<(B + 255) / 256, 256, 0, stream>>>(x, calib, bias, out, B);
    }
}